// GINConvGraphGymLayer_80711025426654
// MI455X (gfx1250) — compile-verified
//
#include <hip/hip_runtime.h>

typedef float v2f __attribute__((ext_vector_type(2)));
typedef float v8f __attribute__((ext_vector_type(8)));

#define NNODES 100000
#define DFEAT  128
#define NEDGES 1600000
#define WTS 130   // LDS stride (floats) for transposed weights: bank-spread, 8B aligned
#define ATS 132   // LDS stride (floats) for activation tiles: 528B = 16B-aligned rows

// generic->LDS address: for the LDS aperture the low 32 bits of a generic
// pointer are exactly the LDS byte offset (ISA 10.2 aperture mapping)
__device__ __forceinline__ unsigned ldsAddr(const void* p) {
    return (unsigned)(size_t)p;
}

// ---------------------------------------------------------------------------
// Kernel 1: h = (1+eps)*x, eps=0  ->  h = x   (h aliases d_out)
// ---------------------------------------------------------------------------
__global__ void gin_init(const float* __restrict__ x, float* __restrict__ h) {
    int t = blockIdx.x * blockDim.x + threadIdx.x;
    if (t < NNODES * DFEAT / 4) {
        ((float4*)h)[t] = ((const float4*)x)[t];
    }
}

// ---------------------------------------------------------------------------
// Kernel 2: scatter-add  h[dst] += x[src]  (one wave32 per edge, float4/lane)
// x + h = 102 MB -> fully L2-resident; atomics resolve at L2.
// ---------------------------------------------------------------------------
__global__ void gin_scatter(const float* __restrict__ x,
                            const long long* __restrict__ ei,
                            float* __restrict__ h) {
    int e    = (blockIdx.x * blockDim.x + threadIdx.x) >> 5;
    int lane = threadIdx.x & 31;
    if (e >= NEDGES) return;
    int src = (int)ei[e];           // row 0 of edge_index
    int dst = (int)ei[NEDGES + e];  // row 1 of edge_index
    int f = lane * 4;
    const float4 v = *(const float4*)(x + (size_t)src * DFEAT + f);
    float* p = h + (size_t)dst * DFEAT + f;
    unsafeAtomicAdd(p + 0, v.x);
    unsafeAtomicAdd(p + 1, v.y);
    unsafeAtomicAdd(p + 2, v.z);
    unsafeAtomicAdd(p + 3, v.w);
}

// ---------------------------------------------------------------------------
// Kernel 3: out = relu(h @ W1 + b1) @ W2 + b2, fp32 WMMA (v_wmma_f32_16x16x4).
// - Weights staged once/block into LDS, TRANSPOSED: B-frag = one ds_load_b64.
// - A tile staged per-wave via global_load_async_to_lds_b128 (coalesced).
// - Result tile streamed out via global_store_async_from_lds_b128.
// ---------------------------------------------------------------------------
__global__ __launch_bounds__(256) void gin_mlp(const float* h,
                                               const float* __restrict__ W1,
                                               const float* __restrict__ b1,
                                               const float* __restrict__ W2,
                                               const float* __restrict__ b2,
                                               float* out) {
    __shared__ __align__(16) float wt1[DFEAT * WTS];   // W1^T, padded
    __shared__ __align__(16) float wt2[DFEAT * WTS];   // W2^T, padded
    __shared__ __align__(16) float act[8 * 16 * ATS];  // per-wave 16x128 tiles

    const int wave = threadIdx.x >> 5;
    const int lane = threadIdx.x & 31;
    const int row  = lane & 15;   // M (A) / N (B,C) index within tile
    const int half = lane >> 4;   // K-half selector (lanes 16-31 carry K+2)
    const int r0   = (blockIdx.x * 8 + wave) * 16;

    // ---- cooperative weight staging (coalesced reads, transposed writes) ----
    for (int i = threadIdx.x; i < DFEAT * DFEAT; i += 256) {
        int k = i >> 7, c = i & 127;
        wt1[c * WTS + k] = W1[i];
        wt2[c * WTS + k] = W2[i];
    }
    __syncthreads();

    if (r0 < NNODES) {
        float* tile = act + wave * 16 * ATS;
        const unsigned tbase = ldsAddr(tile);

        // ---- async-copy this wave's 16x128 A tile into LDS (b128/lane) ----
        const float* gsrc = h + (size_t)r0 * DFEAT;
#pragma unroll
        for (int r = 0; r < 16; ++r) {
            unsigned la = tbase + r * (ATS * 4) + lane * 16;
            unsigned long long ga =
                (unsigned long long)(size_t)(gsrc + r * DFEAT) + (unsigned)(lane * 16);
            asm volatile("global_load_async_to_lds_b128 %0, %1, off"
                         :: "v"(la), "v"(ga) : "memory");
        }
        asm volatile("s_wait_asynccnt 0" ::: "memory");

        // ---------------- Stage 1: C = A(16x128) * W1 ----------------
        v8f c[8];
#pragma unroll
        for (int n = 0; n < 8; ++n) c[n] = (v8f)0.0f;

        for (int k4 = 0; k4 < DFEAT / 4; ++k4) {
            const int kc = k4 * 4 + half * 2;
            v2f a = *(const v2f*)(tile + row * ATS + kc);            // ds_load_b64
#pragma unroll
            for (int n = 0; n < 8; ++n) {
                v2f b = *(const v2f*)(wt1 + (n * 16 + row) * WTS + kc);  // ds_load_b64
                c[n] = __builtin_amdgcn_wmma_f32_16x16x4_f32(
                    false, a, false, b, (short)0, c[n], false, false);
            }
        }

        // bias + ReLU; overwrite tile in place (A fully consumed by now)
#pragma unroll
        for (int n = 0; n < 8; ++n) {
            float bias = b1[n * 16 + row];
#pragma unroll
            for (int r = 0; r < 8; ++r) {
                int m = r + half * 8;   // C layout: VGPR r -> M=r / r+8
                float v = c[n][r] + bias;
                tile[m * ATS + n * 16 + row] = v > 0.0f ? v : 0.0f;
            }
        }

        // ---------------- Stage 2: D = relu(...) * W2 ----------------
        v8f c2[8];
#pragma unroll
        for (int n = 0; n < 8; ++n) c2[n] = (v8f)0.0f;

        for (int k4 = 0; k4 < DFEAT / 4; ++k4) {
            const int kc = k4 * 4 + half * 2;
            v2f a = *(const v2f*)(tile + row * ATS + kc);
#pragma unroll
            for (int n = 0; n < 8; ++n) {
                v2f b = *(const v2f*)(wt2 + (n * 16 + row) * WTS + kc);
                c2[n] = __builtin_amdgcn_wmma_f32_16x16x4_f32(
                    false, a, false, b, (short)0, c2[n], false, false);
            }
        }

        // + b2 into the tile, then stream out coalesced via async store
#pragma unroll
        for (int n = 0; n < 8; ++n) {
            float bias = b2[n * 16 + row];
#pragma unroll
            for (int r = 0; r < 8; ++r) {
                int m = r + half * 8;
                tile[m * ATS + n * 16 + row] = c2[n][r] + bias;
            }
        }
        asm volatile("s_wait_dscnt 0" ::: "memory");  // LDS writes visible to async engine

        float* gdst = out + (size_t)r0 * DFEAT;
#pragma unroll
        for (int r = 0; r < 16; ++r) {
            unsigned la = tbase + r * (ATS * 4) + lane * 16;
            unsigned long long ga =
                (unsigned long long)(size_t)(gdst + r * DFEAT) + (unsigned)(lane * 16);
            asm volatile("global_store_async_from_lds_b128 %0, %1, off"
                         :: "v"(ga), "v"(la) : "memory");
        }
        asm volatile("s_wait_asynccnt 0" ::: "memory");
    }
}

// ---------------------------------------------------------------------------
extern "C" void kernel_launch(void* const* d_in, const int* in_sizes, int n_in,
                              void* d_out, int out_size, void* d_ws, size_t ws_size,
                              hipStream_t stream) {
    const float*     x  = (const float*)d_in[0];
    const long long* ei = (const long long*)d_in[1];  // int64 edge_index [2, E]
    const float*     W1 = (const float*)d_in[2];
    const float*     b1 = (const float*)d_in[3];
    const float*     W2 = (const float*)d_in[4];
    const float*     b2 = (const float*)d_in[5];
    float*           out = (float*)d_out;

    // 1) out <- x
    int initThreads = NNODES * DFEAT / 4;
    gin_init<<<(initThreads + 255) / 256, 256, 0, stream>>>(x, out);

    // 2) out[dst] += x[src]  (one wave per edge)
    long long scatThreads = (long long)NEDGES * 32;
    gin_scatter<<<(unsigned)((scatThreads + 255) / 256), 256, 0, stream>>>(x, ei, out);

    // 3) out <- relu(out @ W1 + b1) @ W2 + b2   (in-place per 16-row tile)
    int tiles = NNODES / 16;               // 6250 tiles, 8 per block
    gin_mlp<<<(tiles + 7) / 8, 256, 0, stream>>>(out, W1, b1, W2, b2, out);
}